// Sparsify_hw_74775380623606
// MI455X (gfx1250) — compile-verified
//
#include <hip/hip_runtime.h>
#include <stdint.h>

// Sparsify_hw: per-(n,c) row of 4096 floats, keep top-25% by |x|, blend by tau.
//   out = x                where |x| in row top-k (k = 1024)
//   out = (1 - tau) * x    otherwise
//
// Strategy (MI455X): one workgroup per row. Async-DMA the 16KB row into LDS
// (CDNA5 GLOBAL_LOAD_ASYNC_TO_LDS_B128, ASYNCcnt), radix-select the k-th
// largest |x| bit-pattern with an LDS histogram (4 x 8-bit passes), then
// apply the threshold and stream results out with 128-bit non-temporal
// stores. Global traffic = read-once + write-once = the 256MB roofline floor.

typedef float v4f __attribute__((ext_vector_type(4)));

#define HW    4096
#define KSEL  1024
#define NTHR  256

__device__ __forceinline__ void async_copy_b128(uint32_t lds_off, uint64_t gaddr) {
  // CDNA5 async global->LDS, tracked with ASYNCcnt (CDNA5 ISA).
  asm volatile("global_load_async_to_lds_b128 %0, %1, off"
               :: "v"(lds_off), "v"(gaddr) : "memory");
}

__device__ __forceinline__ void wait_async0() {
#if __has_builtin(__builtin_amdgcn_s_wait_asynccnt)
  __builtin_amdgcn_s_wait_asynccnt(0);
#else
  asm volatile("s_wait_asynccnt 0" ::: "memory");
#endif
}

__global__ __launch_bounds__(NTHR) void sparsify_topk_kernel(
    const float* __restrict__ x, const float* __restrict__ tau_p,
    float* __restrict__ out)
{
  __shared__ float    s_x[HW];          // 16 KB row staging
  __shared__ uint32_t s_hist[256];      // radix histogram
  __shared__ uint32_t s_bin, s_above, s_take;

  const int      tid  = threadIdx.x;
  const uint64_t base = (uint64_t)blockIdx.x * (uint64_t)HW;

  // ---- Stage the row into LDS with async b128 copies (4 per thread) ----
  const uint32_t lds_base = (uint32_t)(uintptr_t)&s_x[0];   // low 32b = LDS byte offset
  const uint64_t g_base   = (uint64_t)(uintptr_t)(x + base);
#pragma unroll
  for (int j = 0; j < HW / (4 * NTHR); ++j) {
    const uint32_t off = (uint32_t)(tid + j * NTHR) * 16u;  // bytes, coalesced
    async_copy_b128(lds_base + off, g_base + off);
  }
  wait_async0();        // each wave drains its own ASYNCcnt
  __syncthreads();      // all 8 waves' data visible in LDS

  // ---- Radix-select the KSEL-th largest key (key = bits of |x|) ----
  uint32_t prefix    = 0;
  uint32_t remaining = KSEL;
#pragma unroll
  for (int shift = 24; shift >= 0; shift -= 8) {
    for (int b = tid; b < 256; b += NTHR) s_hist[b] = 0;
    __syncthreads();
    const uint32_t pmask = (shift == 24) ? 0u : (0xFFFFFFFFu << (shift + 8));
    for (int i = tid; i < HW; i += NTHR) {
      const uint32_t key = __float_as_uint(s_x[i]) & 0x7FFFFFFFu;
      if ((key & pmask) == prefix)
        atomicAdd(&s_hist[(key >> shift) & 0xFFu], 1u);
    }
    __syncthreads();
    if (tid == 0) {
      uint32_t cum = 0, b = 255;
      for (;;) {                      // scan bins high->low for the k-th rank
        const uint32_t c = s_hist[b];
        if (cum + c >= remaining || b == 0) break;
        cum += c;
        --b;
      }
      s_bin   = b;
      s_above = cum;                  // strictly-above count within this prefix
    }
    __syncthreads();
    prefix    |= (s_bin << shift);
    remaining -= s_above;
  }

  // prefix = threshold key T; remaining = how many keys == T belong in top-k.
  const uint32_t T     = prefix;
  const uint32_t quota = remaining;
  if (tid == 0) s_take = 0;
  const float tau   = *tau_p;
  const float scale = 1.0f - tau;
  __syncthreads();

  // ---- Apply threshold, write out with 128-bit non-temporal stores ----
  const v4f* s_x4 = (const v4f*)s_x;
  v4f*       o4   = (v4f*)(out + base);
#pragma unroll
  for (int j = 0; j < HW / (4 * NTHR); ++j) {
    const int v = tid + j * NTHR;
    v4f val = s_x4[v];                // ds_load_b128
#pragma unroll
    for (int e = 0; e < 4; ++e) {
      const float    f   = val[e];
      const uint32_t key = __float_as_uint(f) & 0x7FFFFFFFu;
      if (key < T) {
        val[e] = f * scale;
      } else if (key == T) {          // ties: fill quota (order-agnostic)
        if (atomicAdd(&s_take, 1u) >= quota) val[e] = f * scale;
      }
    }
    __builtin_nontemporal_store(val, o4 + v);   // write-once: bypass L2 dirty
  }
}

extern "C" void kernel_launch(void* const* d_in, const int* in_sizes, int n_in,
                              void* d_out, int out_size, void* d_ws, size_t ws_size,
                              hipStream_t stream) {
  const float* x   = (const float*)d_in[0];
  const float* tau = (const float*)d_in[1];
  float*       out = (float*)d_out;
  const int rows = in_sizes[0] / HW;   // n*c = 8192
  sparsify_topk_kernel<<<rows, NTHR, 0, stream>>>(x, tau, out);
}